// TriplaneRefinementLayer_42666205118752
// MI455X (gfx1250) — compile-verified
//
#include <hip/hip_runtime.h>
#include <hip/hip_bf16.h>

// Triplane refinement, exploiting the separability of the sampling grid:
//   ctx[i,j] = (Mcol[j] + Mrow[i]) / (2S)   per plane,
// so the K/V projections reduce to 12 GEMMs of 192x64 @ 64x64 done with
// V_WMMA_F32_16X16X4_F32, followed by one memory-bound fused kernel.

#define TGRID 192
#define TC    64
#define TS    16
#define THW   (TGRID * TGRID)

typedef __attribute__((ext_vector_type(2))) float v2f;
typedef __attribute__((ext_vector_type(8))) float v8f;

// ---------------------------------------------------------------------------
// Phase 1: build the 6 separable context matrices Smat[matIdx][n][c]
//   matIdx: 0 = xy/col(j)  src=xz  (cx=u[n], cy=lin(sb4,sb5))
//           1 = xy/row(i)  src=yz  (cx=u[n], cy=lin(sb4,sb5))
//           2 = xz/col(j)  src=xy  (cx=u[n], cy=lin(sb2,sb3))
//           3 = xz/row(i)  src=yz  (cx=lin(sb2,sb3), cy=u[n])
//           4 = yz/col(j)  src=xy  (cx=lin(sb0,sb1), cy=u[n])
//           5 = yz/row(i)  src=xz  (cx=lin(sb0,sb1), cy=u[n])
// Each entry = sum over S bilinear taps (mean's 1/(2S) applied later).
// ---------------------------------------------------------------------------
__global__ __launch_bounds__(64) void triplane_sample_kernel(
    const float* __restrict__ xy, const float* __restrict__ xz,
    const float* __restrict__ yz, const float* __restrict__ sb,
    float* __restrict__ Smat)
{
  const int blk    = blockIdx.x;          // 6*192 blocks
  const int matIdx = blk / TGRID;
  const int n      = blk - matIdx * TGRID;
  const int c      = threadIdx.x;         // 64 channels

  const float* src; float lo, hi; int swap;
  switch (matIdx) {
    case 0:  src = xz; lo = sb[4]; hi = sb[5]; swap = 0; break;
    case 1:  src = yz; lo = sb[4]; hi = sb[5]; swap = 0; break;
    case 2:  src = xy; lo = sb[2]; hi = sb[3]; swap = 0; break;
    case 3:  src = yz; lo = sb[2]; hi = sb[3]; swap = 1; break;
    case 4:  src = xy; lo = sb[0]; hi = sb[1]; swap = 1; break;
    default: src = xz; lo = sb[0]; hi = sb[1]; swap = 1; break;
  }

  const float un = -1.0f + (2.0f / (TGRID - 1)) * (float)n;
  const float* f = src + c * THW;
  float acc = 0.0f;

  for (int s = 0; s < TS; ++s) {
    const float ls = lo + (hi - lo) * ((float)s * (1.0f / (TS - 1)));
    const float cx = swap ? ls : un;
    const float cy = swap ? un : ls;
    float x = fminf(fmaxf((cx + 1.0f) * 0.5f * (TGRID - 1), 0.0f), (float)(TGRID - 1));
    float y = fminf(fmaxf((cy + 1.0f) * 0.5f * (TGRID - 1), 0.0f), (float)(TGRID - 1));
    const float x0f = floorf(x), y0f = floorf(y);
    const float wx = x - x0f, wy = y - y0f;
    const int x0 = (int)x0f, y0 = (int)y0f;
    const int x1 = min(x0 + 1, TGRID - 1);
    const int y1 = min(y0 + 1, TGRID - 1);
    const float f00 = f[y0 * TGRID + x0];
    const float f01 = f[y0 * TGRID + x1];
    const float f10 = f[y1 * TGRID + x0];
    const float f11 = f[y1 * TGRID + x1];
    acc += (f00 * (1.0f - wx) + f01 * wx) * (1.0f - wy) +
           (f10 * (1.0f - wx) + f11 * wx) * wy;
  }
  Smat[matIdx * (TGRID * TC) + n * TC + c] = acc;   // row-major [n][c] (A matrix)
}

// ---------------------------------------------------------------------------
// Phase 2: 12 GEMMs (6 matrices x {k_w, v_w}): D(192x64) = A(192x64) @ W^T.
// One wave per 16x16 output tile, K=64 -> 16 chained v_wmma_f32_16x16x4_f32.
// 576 tile-jobs = 72 blocks * 8 wave32s. EXEC is all-1s (no divergence).
// A-operand layout (ISA 7.12.2, 32-bit A 16x4): lanes 0-15 hold K={0,1},
// lanes 16-31 hold K={2,3}; B mirrors with M<->N. D: VGPR r -> row r (+8 for
// upper half-wave), col = lane&15. Output stored transposed [c][n] so the
// fused kernel reads it coalesced.
// ---------------------------------------------------------------------------
__global__ __launch_bounds__(256) void triplane_gemm_kernel(
    const float* __restrict__ Smat, const float* __restrict__ k_w,
    const float* __restrict__ v_w, float* __restrict__ KV)
{
  const int lane = threadIdx.x & 31;
  const int wave = (int)((blockIdx.x * blockDim.x + threadIdx.x) >> 5); // 0..575
  const int gemm = wave / 48;          // 0..11
  const int tile = wave - gemm * 48;   // 0..47
  const int rowTile = tile >> 2;       // 0..11  (rows of A, n index)
  const int colTile = tile & 3;        // 0..3   (output channels)
  const int matIdx  = gemm >> 1;       // 0..5
  const int wsel    = gemm & 1;        // 0 = k_w, 1 = v_w

  const float* A = Smat + matIdx * (TGRID * TC);   // [192][64]
  const float* W = wsel ? v_w : k_w;               // [o][c] row-major; B[k][n] = W[n][k]

  const int half = (lane < 16) ? 0 : 2;
  const int l15  = lane & 15;
  const int arow = rowTile * 16 + l15;             // A row (n)
  const int bcol = colTile * 16 + l15;             // output channel (o)

  v8f acc = {0.f, 0.f, 0.f, 0.f, 0.f, 0.f, 0.f, 0.f};
  #pragma unroll
  for (int kk = 0; kk < 16; ++kk) {
    const int kb = kk * 4 + half;
    v2f a, b;
    a.x = A[arow * TC + kb];     a.y = A[arow * TC + kb + 1];
    b.x = W[bcol * TC + kb];     b.y = W[bcol * TC + kb + 1];
    acc = __builtin_amdgcn_wmma_f32_16x16x4_f32(
        /*neg_a=*/false, a, /*neg_b=*/false, b,
        /*c_mod=*/(short)0, acc, /*reuse_a=*/false, /*reuse_b=*/false);
  }

  float* out = KV + (matIdx * 2 + wsel) * (TGRID * TC);  // layout [c][n]
  const int mbase = rowTile * 16 + ((lane < 16) ? 0 : 8);
  #pragma unroll
  for (int r = 0; r < 8; ++r)
    out[bcol * TGRID + (mbase + r)] = acc[r];
}

// ---------------------------------------------------------------------------
// Phase 3 (the memory-bound part): one thread per pixel.
//   qn = layernorm(qf); k = (Kcol[j]+Krow[i])/32 + k_b; v likewise;
//   out = qf + qn*k*v,  written in (plane, C, H, W) layout.
// Per-channel global reads/writes are stride-1 across lanes (128B/wave).
// ---------------------------------------------------------------------------
__global__ __launch_bounds__(256) void triplane_fuse_kernel(
    const float* __restrict__ xy, const float* __restrict__ xz,
    const float* __restrict__ yz,
    const float* __restrict__ ln_g, const float* __restrict__ ln_b,
    const float* __restrict__ k_b, const float* __restrict__ v_b,
    const float* __restrict__ KV, float* __restrict__ out)
{
  const int gid = blockIdx.x * blockDim.x + threadIdx.x;  // 0 .. 3*HW-1
  const int p   = gid / THW;
  const int pix = gid - p * THW;
  const int i   = pix / TGRID;
  const int j   = pix - i * TGRID;

  const float* plane = (p == 0) ? xy : ((p == 1) ? xz : yz);
  const float* Kcol = KV + ((2 * p    ) * 2 + 0) * (TGRID * TC);
  const float* Vcol = KV + ((2 * p    ) * 2 + 1) * (TGRID * TC);
  const float* Krow = KV + ((2 * p + 1) * 2 + 0) * (TGRID * TC);
  const float* Vrow = KV + ((2 * p + 1) * 2 + 1) * (TGRID * TC);

  float qf[TC];
  float sum = 0.0f;
  #pragma unroll
  for (int c = 0; c < TC; ++c) {
    qf[c] = plane[c * THW + pix];
    sum += qf[c];
  }
  const float m = sum * (1.0f / TC);
  float var = 0.0f;
  #pragma unroll
  for (int c = 0; c < TC; ++c) {
    const float d = qf[c] - m;
    var += d * d;
  }
  var *= (1.0f / TC);
  const float inv = rsqrtf(var + 1e-5f);

  float* o = out + p * (TC * THW) + pix;
  #pragma unroll
  for (int c = 0; c < TC; ++c) {
    const float qn = (qf[c] - m) * inv * ln_g[c] + ln_b[c];
    const float kk = (Kcol[c * TGRID + j] + Krow[c * TGRID + i]) * (1.0f / 32.0f) + k_b[c];
    const float vv = (Vcol[c * TGRID + j] + Vrow[c * TGRID + i]) * (1.0f / 32.0f) + v_b[c];
    o[c * THW] = qf[c] + qn * kk * vv;
  }
}

// ---------------------------------------------------------------------------
extern "C" void kernel_launch(void* const* d_in, const int* in_sizes, int n_in,
                              void* d_out, int out_size, void* d_ws, size_t ws_size,
                              hipStream_t stream)
{
  (void)in_sizes; (void)n_in; (void)out_size; (void)ws_size;
  const float* xy   = (const float*)d_in[0];
  const float* xz   = (const float*)d_in[1];
  const float* yz   = (const float*)d_in[2];
  // d_in[3..6] (image_features, depth, c2w, K) are unused by the reference.
  const float* sb   = (const float*)d_in[7];
  const float* ln_g = (const float*)d_in[8];
  const float* ln_b = (const float*)d_in[9];
  const float* k_w  = (const float*)d_in[10];
  const float* k_b  = (const float*)d_in[11];
  const float* v_w  = (const float*)d_in[12];
  const float* v_b  = (const float*)d_in[13];

  float* Smat = (float*)d_ws;                 // 6 * 192 * 64 floats
  float* KV   = Smat + 6 * TGRID * TC;        // 12 * 192 * 64 floats
  float* out  = (float*)d_out;                // 3 * 64 * 192 * 192 floats

  triplane_sample_kernel<<<6 * TGRID, 64, 0, stream>>>(xy, xz, yz, sb, Smat);
  triplane_gemm_kernel<<<72, 256, 0, stream>>>(Smat, k_w, v_w, KV);
  triplane_fuse_kernel<<<(3 * THW) / 256, 256, 0, stream>>>(
      xy, xz, yz, ln_g, ln_b, k_b, v_b, KV, out);
}